// EntropyAttentionHead_90048284328255
// MI455X (gfx1250) — compile-verified
//
#include <hip/hip_runtime.h>
#include <stdint.h>

// ---------------------------------------------------------------------------
// EntropyAttentionHead: per-(b,c) 256-bin histogram -> Shannon entropy ->
// broadcast. Bandwidth-bound: 411 MB read + 411 MB write @ 23.3 TB/s ~ 35 us.
// No matmul structure -> WMMA not applicable; the CDNA5 feature that fits is
// the async global->LDS data mover (ASYNCcnt), used as a 4-deep per-wave
// prefetch queue feeding wave-private LDS histograms (ds_add_u32, exact).
// ---------------------------------------------------------------------------

#define TPB    256         // 8 wave32s per block
#define WAVES  8
#define BINS   256
#define N_PIX  50176       // 224*224
#define ITERS  49          // 50176 / (256 threads * 4 floats)
#define STAGES 4           // async pipeline depth per wave

typedef float __attribute__((ext_vector_type(4))) floatx4;

// Issue an async 16B-per-lane copy global -> LDS (tracked by ASYNCcnt).
__device__ __forceinline__ void async_ld_b128(uint32_t lds_off, const void* gptr) {
  asm volatile("global_load_async_to_lds_b128 %0, %1, off"
               :: "v"(lds_off), "v"(gptr)
               : "memory");
}

// Generic pointers to LDS carry the LDS byte offset in their low 32 bits
// (flat-aperture rule).
__device__ __forceinline__ uint32_t lds_off_of(const void* p) {
  return (uint32_t)(uintptr_t)p;
}

__global__ void __launch_bounds__(TPB)
entropy_hist_kernel(const float* __restrict__ x, float* __restrict__ out) {
  __shared__ uint32_t hist[WAVES][BINS];          // 8 KB  per-wave histograms
  __shared__ float    stage[WAVES][STAGES][128];  // 16 KB 4 x 512B queue / wave
  __shared__ float    red[TPB];                   // 1 KB  entropy reduction

  const int tid  = threadIdx.x;
  const int wave = tid >> 5;                      // wave32
  const int lane = tid & 31;
  const int bc   = blockIdx.x;                    // (b*C + c)

  #pragma unroll
  for (int w = 0; w < WAVES; ++w) hist[w][tid] = 0u;
  __syncthreads();

  // Each thread streams 49 float4s; a wave's 32 lanes cover a contiguous
  // 512B slice per tile, staged through the wave's 4-deep LDS queue.
  const float*   gsrc  = x + (size_t)bc * N_PIX + (size_t)tid * 4;
  const float*   sbase = &stage[wave][0][lane * 4];
  const uint32_t lbase = lds_off_of(sbase);       // buffer j at lbase + (j<<9)

  // Prologue: tiles 0..2 -> buffers 0..2 (3 outstanding).
  async_ld_b128(lbase,           gsrc);
  async_ld_b128(lbase + (1u<<9), gsrc + 1 * (TPB * 4));
  async_ld_b128(lbase + (2u<<9), gsrc + 2 * (TPB * 4));

  for (int i = 0; i < ITERS; ++i) {
    if (i + 3 < ITERS) {
      // WAR guard: ds_reads of the buffer being overwritten (consumed last
      // iteration) must have drained before the async engine rewrites it.
      asm volatile("s_wait_dscnt 0" ::: "memory");
      async_ld_b128(lbase + ((uint32_t)((i + 3) & 3) << 9),
                    gsrc + (size_t)(i + 3) * (TPB * 4));
      asm volatile("s_wait_asynccnt 3" ::: "memory");  // retire tile i
    } else if (i + 2 < ITERS) {
      asm volatile("s_wait_asynccnt 2" ::: "memory");
    } else if (i + 1 < ITERS) {
      asm volatile("s_wait_asynccnt 1" ::: "memory");
    } else {
      asm volatile("s_wait_asynccnt 0" ::: "memory");
    }

    // Single ds_load_b128 of this thread's staged 16B.
    const floatx4 v = *(const floatx4*)(sbase + (size_t)(i & 3) * 128);

    #pragma unroll
    for (int k = 0; k < 4; ++k) {
      const float f = v[k];
      int b = (int)floorf(f * 256.0f);
      b = b < 0 ? 0 : (b > (BINS - 1) ? (BINS - 1) : b);
      if (f >= 0.0f && f <= 1.0f)                  // torch.histc validity
        atomicAdd(&hist[wave][b], 1u);             // ds_add_u32 (exact int)
    }
  }
  __syncthreads();

  // Thread t owns bin t: merge the 8 wave-private counts, entropy term.
  uint32_t h = 0;
  #pragma unroll
  for (int w = 0; w < WAVES; ++w) h += hist[w][tid];
  const float p    = (float)h * (1.0f / (float)N_PIX);
  const float term = -(p * logf(p + 1e-10f));

  // Deterministic tree reduction over the 256 bin terms.
  red[tid] = term;
  __syncthreads();
  for (int s = TPB / 2; s > 0; s >>= 1) {
    if (tid < s) red[tid] += red[tid + s];
    __syncthreads();
  }
  const float ent = fabsf(red[0]);

  // Broadcast the scalar over the (b,c) spatial map with streaming NT stores
  // (write-once data: keep it out of L2's working set).
  const floatx4 o = {ent, ent, ent, ent};
  float* obase = out + (size_t)bc * N_PIX + (size_t)tid * 4;
  #pragma unroll 4
  for (int i = 0; i < ITERS; ++i)
    __builtin_nontemporal_store(o, (floatx4*)(obase + (size_t)i * (TPB * 4)));
}

extern "C" void kernel_launch(void* const* d_in, const int* in_sizes, int n_in,
                              void* d_out, int out_size, void* d_ws, size_t ws_size,
                              hipStream_t stream) {
  const float* x   = (const float*)d_in[0];   // [B,C,H,W] float32
  float*       out = (float*)d_out;           // [B,C,H,W] float32
  const int blocks = in_sizes[0] / N_PIX;     // B*C = 2048
  entropy_hist_kernel<<<blocks, TPB, 0, stream>>>(x, out);
}